// TemporalContrastiveModel_10264971837866
// MI455X (gfx1250) — compile-verified
//
#include <hip/hip_runtime.h>

typedef float v2f __attribute__((ext_vector_type(2)));
typedef float v8f __attribute__((ext_vector_type(8)));

#define TPB 256

// ---------------------------------------------------------------- utilities
__global__ void k_fill(float* __restrict__ p, float v, int n) {
  int i = blockIdx.x * blockDim.x + threadIdx.x;
  if (i < n) p[i] = v;
}

__global__ void k_deg_accum(const int* __restrict__ dst, float* __restrict__ deg, int E) {
  int e = blockIdx.x * blockDim.x + threadIdx.x;
  if (e < E) atomicAdd(&deg[dst[e]], 1.0f);
}

__global__ void k_rsqrt(float* __restrict__ d, int n) {
  int i = blockIdx.x * blockDim.x + threadIdx.x;
  if (i < n) d[i] = rsqrtf(d[i]);   // deg >= 1 always (self-loops)
}

// ------------------------------------------------- WMMA f32 GEMM  C = A(K) x W(K,64)
// One wave computes a 16x64 output tile using V_WMMA_F32_16X16X4_F32.
// A fragment (16x4): lanes 0-15 hold M=lane, K=koff..koff+1 with koff=0;
//                    lanes 16-31 hold M=lane-16, koff=2.     (ISA 7.12.2)
// B fragment (4x16): VGPR0 = row K=koff, VGPR1 = row K=koff+1, N = lane&15.
// D (16x16 f32):     VGPR i -> row i (lanes 0-15) / row i+8 (lanes 16-31), N = lane&15.
template <int K, int BIAS, int RELU>
__global__ __launch_bounds__(TPB) void k_gemm_wmma(const float* __restrict__ A,
                                                   const float* __restrict__ W,
                                                   const float* __restrict__ bias,
                                                   float* __restrict__ C, int nRows) {
  __shared__ float sW[K * 64];
  for (int i = threadIdx.x; i < K * 64; i += TPB) sW[i] = W[i];
  __syncthreads();

  const int lane = threadIdx.x & 31;
  const int wave = threadIdx.x >> 5;
  const int tile = blockIdx.x * (TPB / 32) + wave;
  const int rowBase = tile * 16;
  if (rowBase >= nRows) return;                 // wave-uniform exit (EXEC all-1 for WMMA)

  const int m    = lane & 15;                   // M (for A) / N (for B,D) within tile
  const int koff = (lane >> 4) << 1;            // 0 for lanes 0-15, 2 for lanes 16-31
  const float* __restrict__ Arow = A + (size_t)(rowBase + m) * K;

  v8f acc0 = {}, acc1 = {}, acc2 = {}, acc3 = {};
#pragma unroll 4
  for (int k = 0; k < K; k += 4) {
    v2f a;
    a.x = Arow[k + koff];
    a.y = Arow[k + koff + 1];
    const float* w0 = &sW[(k + koff) * 64];
    const float* w1 = &sW[(k + koff + 1) * 64];
    v2f b0, b1, b2, b3;
    b0.x = w0[m];      b0.y = w1[m];
    b1.x = w0[16 + m]; b1.y = w1[16 + m];
    b2.x = w0[32 + m]; b2.y = w1[32 + m];
    b3.x = w0[48 + m]; b3.y = w1[48 + m];
    acc0 = __builtin_amdgcn_wmma_f32_16x16x4_f32(false, a, false, b0, (short)0, acc0, false, false);
    acc1 = __builtin_amdgcn_wmma_f32_16x16x4_f32(false, a, false, b1, (short)0, acc1, false, false);
    acc2 = __builtin_amdgcn_wmma_f32_16x16x4_f32(false, a, false, b2, (short)0, acc2, false, false);
    acc3 = __builtin_amdgcn_wmma_f32_16x16x4_f32(false, a, false, b3, (short)0, acc3, false, false);
  }

  float bv0 = 0.f, bv1 = 0.f, bv2 = 0.f, bv3 = 0.f;
  if (BIAS) { bv0 = bias[m]; bv1 = bias[16 + m]; bv2 = bias[32 + m]; bv3 = bias[48 + m]; }

  const int rOff = (lane >> 4) * 8;             // D layout: VGPR i -> row i or i+8
#pragma unroll
  for (int i = 0; i < 8; ++i) {
    int row = rowBase + rOff + i;
    if (row < nRows) {
      float v0 = acc0[i] + bv0, v1 = acc1[i] + bv1, v2 = acc2[i] + bv2, v3 = acc3[i] + bv3;
      if (RELU) {
        v0 = fmaxf(v0, 0.f); v1 = fmaxf(v1, 0.f); v2 = fmaxf(v2, 0.f); v3 = fmaxf(v3, 0.f);
      }
      float* __restrict__ Crow = C + (size_t)row * 64;
      Crow[m] = v0; Crow[16 + m] = v1; Crow[32 + m] = v2; Crow[48 + m] = v3;
    }
  }
}

// ---------------------------------------------------------- edge scatter-add
// one thread per (edge, dim): agg[dst][d] += h[src][d] * dinv[src]*dinv[dst]
__global__ __launch_bounds__(TPB) void k_edge_agg(const int* __restrict__ src,
                                                  const int* __restrict__ dst,
                                                  const float* __restrict__ dinv,
                                                  const float* __restrict__ h,
                                                  float* __restrict__ agg, int E) {
  long long t = (long long)blockIdx.x * blockDim.x + threadIdx.x;
  long long e = t >> 6;
  int d = (int)(t & 63);
  if (e >= (long long)E) return;
  int s = src[e], q = dst[e];
  float nrm = dinv[s] * dinv[q];
  atomicAdd(&agg[(size_t)q * 64 + d], h[(size_t)s * 64 + d] * nrm);
}

// ----------------------------------------------- epilogue: + self-loop + bias (+relu)
template <int RELU>
__global__ __launch_bounds__(TPB) void k_epilogue(const float* __restrict__ agg,
                                                  const float* __restrict__ hself,
                                                  const float* __restrict__ dinv,
                                                  const float* __restrict__ bias,
                                                  float* __restrict__ out, int n) {
  int t = blockIdx.x * blockDim.x + threadIdx.x;
  if (t >= n * 64) return;
  int node = t >> 6, d = t & 63;
  float di = dinv[node];
  float v = agg[t] + hself[t] * di * di + bias[d];
  if (RELU) v = fmaxf(v, 0.f);
  out[t] = v;
}

// --------------------------------------------------------------------- host
extern "C" void kernel_launch(void* const* d_in, const int* in_sizes, int n_in,
                              void* d_out, int out_size, void* d_ws, size_t ws_size,
                              hipStream_t stream) {
  const float* x   = (const float*)d_in[0];
  const int*   ei  = (const int*)d_in[1];
  const float* W1  = (const float*)d_in[2];
  const float* b1  = (const float*)d_in[3];
  const float* W2  = (const float*)d_in[4];
  const float* b2  = (const float*)d_in[5];
  const float* Wp1 = (const float*)d_in[6];
  const float* bp1 = (const float*)d_in[7];
  const float* Wp2 = (const float*)d_in[8];
  const float* bp2 = (const float*)d_in[9];

  const int N = in_sizes[0] / 128;
  const int E = in_sizes[1] / 2;
  const int* src = ei;
  const int* dst = ei + E;

  float* z = (float*)d_out;                     // output 0: [N,64]
  float* p = (float*)d_out + (size_t)N * 64;    // output 1: [N,64]

  float* ws   = (float*)d_ws;
  float* dinv = ws;                             // N   (deg -> dinv)
  float* bufA = ws + N;                         // N*64 (h0, later h1)
  float* bufB = bufA + (size_t)N * 64;          // N*64 (agg1 -> h -> r)

  const int nd = N * 64;
  const int tiles = (N + 15) / 16;
  const int gblk  = (tiles + (TPB / 32) - 1) / (TPB / 32);
  const unsigned eblk = (unsigned)(((long long)E * 64 + TPB - 1) / TPB);

  // degrees: deg = 1 (self-loop) + in-edges;  dinv = rsqrt(deg)
  k_fill<<<(N + TPB - 1) / TPB, TPB, 0, stream>>>(dinv, 1.0f, N);
  k_deg_accum<<<(E + TPB - 1) / TPB, TPB, 0, stream>>>(dst, dinv, E);
  k_rsqrt<<<(N + TPB - 1) / TPB, TPB, 0, stream>>>(dinv, N);

  // layer 1: h0 = x @ W1
  k_gemm_wmma<128, 0, 0><<<gblk, TPB, 0, stream>>>(x, W1, nullptr, bufA, N);
  k_fill<<<(nd + TPB - 1) / TPB, TPB, 0, stream>>>(bufB, 0.0f, nd);
  k_edge_agg<<<eblk, TPB, 0, stream>>>(src, dst, dinv, bufA, bufB, E);
  k_epilogue<1><<<(nd + TPB - 1) / TPB, TPB, 0, stream>>>(bufB, bufA, dinv, b1, bufB, N);

  // layer 2: h1 = h @ W2; z = agg2 + self + b2 (written into d_out)
  k_gemm_wmma<64, 0, 0><<<gblk, TPB, 0, stream>>>(bufB, W2, nullptr, bufA, N);
  k_fill<<<(nd + TPB - 1) / TPB, TPB, 0, stream>>>(z, 0.0f, nd);
  k_edge_agg<<<eblk, TPB, 0, stream>>>(src, dst, dinv, bufA, z, E);
  k_epilogue<0><<<(nd + TPB - 1) / TPB, TPB, 0, stream>>>(z, bufA, dinv, b2, z, N);

  // MLP head: r = relu(z@Wp1 + bp1); p = r@Wp2 + bp2
  k_gemm_wmma<64, 1, 1><<<gblk, TPB, 0, stream>>>(z, Wp1, bp1, bufB, N);
  k_gemm_wmma<64, 1, 0><<<gblk, TPB, 0, stream>>>(bufB, Wp2, bp2, p, N);
}